// DistanceTransformLoss_87969520157281
// MI455X (gfx1250) — compile-verified
//
#include <hip/hip_runtime.h>
#include <math.h>

// Problem geometry (fixed by the reference): (N=32, C=1, H=1024, W=1024) f32.
#define NH    1024
#define NW    1024
#define NIMG  32
#define TPB   64          // columns per block (2 waves; each wave owns 32 columns)
#define CHUNK 16          // rows staged per async buffer
#define NCHUNK (NH / CHUNK)

#define DIST_BYTES  (NH / 2 * TPB * 4)    // 128 KB: u32 = two packed u16 rows
#define STAGE_BYTES (CHUNK * TPB * 4)     // 4 KB per staging buffer
#define NBUF        3                      // triple buffering (2 chunks in flight)
#define LDS_TOTAL   (DIST_BYTES + 2 * NBUF * STAGE_BYTES)  // 155648 B (<160KB -> 2 blocks/WGP)

// Wave-wide async global->LDS copy: each lane moves 16 contiguous bytes.
__device__ __forceinline__ void async_g2l_b128(unsigned lds_byte_off,
                                               unsigned global_byte_off,
                                               unsigned long long base) {
  asm volatile("global_load_async_to_lds_b128 %0, %1, %2"
               :
               : "v"(lds_byte_off), "v"(global_byte_off), "s"(base)
               : "memory");
}
__device__ __forceinline__ void wait_async16() { asm volatile("s_wait_asynccnt 0x10" ::: "memory"); }
__device__ __forceinline__ void wait_async8()  { asm volatile("s_wait_asynccnt 0x8"  ::: "memory"); }
__device__ __forceinline__ void wait_async0()  { asm volatile("s_wait_asynccnt 0x0"  ::: "memory"); }

struct __align__(16) RedBlk { float bce; float tot; unsigned cnt; unsigned pad; };

__global__ __launch_bounds__(TPB) void dtl_main(const float* __restrict__ preds,
                                                const float* __restrict__ tgts,
                                                float* __restrict__ acc,      // [0]=bce_sum [1]=penalty_sum
                                                unsigned* __restrict__ cnt) { // penalty!=0 count
  __shared__ RedBlk s_red;                  // 16 B static -> dynamic LDS base stays 16B aligned
  extern __shared__ unsigned char smem_raw[];

  const int tid   = threadIdx.x;
  const int b     = blockIdx.x;
  const int n     = b >> 4;                 // 16 blocks per image (1024/64)
  const int wbase = (b & 15) * TPB;
  const size_t colbase = (size_t)n * (size_t)(NH * NW) + (size_t)wbase;

  // Low 32 bits of a generic pointer into LDS == byte offset in the LDS aperture.
  const unsigned lds0    = (unsigned)(size_t)(void*)smem_raw;
  const unsigned stTBase = lds0 + DIST_BYTES;
  const unsigned stPBase = stTBase + NBUF * STAGE_BYTES;

  const unsigned long long tg = (unsigned long long)(const void*)(tgts  + colbase);
  const unsigned long long pg = (unsigned long long)(const void*)(preds + colbase);

  // b128 lane mapping: wave wv covers columns [32*wv, 32*wv+32).
  // Lane l: rq = l>>3 selects row-within-group-of-4, q = l&7 selects 16B quad.
  const unsigned wv = (unsigned)tid >> 5;
  const unsigned l  = (unsigned)tid & 31u;
  const unsigned rq = l >> 3, q = l & 7u;
  const unsigned laneLds = rq * (TPB * 4u) + wv * 128u + q * 16u;  // row-major [r][col]*4B
  const unsigned laneG   = rq * (NW * 4u)  + wv * 128u + q * 16u;

  auto issue_chunk = [&](int c, int buf) {
    const unsigned lT = stTBase + (unsigned)buf * STAGE_BYTES + laneLds;
    const unsigned lP = stPBase + (unsigned)buf * STAGE_BYTES + laneLds;
    const unsigned g  = (unsigned)c * (CHUNK * NW * 4u) + laneG;
#pragma unroll
    for (int i = 0; i < CHUNK / 4; ++i) {   // each b128 instr stages 4 rows/wave
      async_g2l_b128(lT + (unsigned)i * (4u * TPB * 4u), g + (unsigned)i * (4u * NW * 4u), tg);
      async_g2l_b128(lP + (unsigned)i * (4u * TPB * 4u), g + (unsigned)i * (4u * NW * 4u), pg);
    }
  };

  unsigned* dist32 = (unsigned*)smem_raw;   // [NH/2][TPB] packed u16 pairs

  // ================= downward pass: BCE + forward min-plus distance =============
  float bce   = 0.0f;
  int   carry = 1 << 20;                    // "infinity" carry-in

  issue_chunk(0, 0);
  issue_chunk(1, 1);
  for (int c = 0; c < NCHUNK; ++c) {
    const int buf = c % NBUF;
    if (c + 2 < NCHUNK) { issue_chunk(c + 2, (c + 2) % NBUF); wait_async16(); }
    else if (c + 1 < NCHUNK) { wait_async8(); }
    else                     { wait_async0(); }

    const float* sT = (const float*)(smem_raw + DIST_BYTES + buf * STAGE_BYTES);
    const float* sP = (const float*)(smem_raw + DIST_BYTES + NBUF * STAGE_BYTES + buf * STAGE_BYTES);
#pragma unroll
    for (int r = 0; r < CHUNK; r += 2) {
      const float t0 = sT[(r + 0) * TPB + tid];
      const float p0 = sP[(r + 0) * TPB + tid];
      const float t1 = sT[(r + 1) * TPB + tid];
      const float p1 = sP[(r + 1) * TPB + tid];
      // stable BCE-with-logits: max(p,0) - p*t + log(1+exp(-|p|))
      bce += fmaxf(p0, 0.0f) - p0 * t0 + __logf(1.0f + __expf(-fabsf(p0)));
      bce += fmaxf(p1, 0.0f) - p1 * t1 + __logf(1.0f + __expf(-fabsf(p1)));
      // forward grassfire: d = (t==1) ? 0 : min(carry+1, H)
      const int d0 = (t0 == 1.0f) ? 0 : min(carry + 1, NH);
      const unsigned v0 = (unsigned)d0 | (p0 > 0.0f ? 0x8000u : 0u);
      const int d1 = (t1 == 1.0f) ? 0 : min(d0 + 1, NH);
      const unsigned v1 = (unsigned)d1 | (p1 > 0.0f ? 0x8000u : 0u);
      carry = d1;
      dist32[((c * CHUNK + r) >> 1) * TPB + tid] = v0 | (v1 << 16);
    }
  }

  // ================= upward pass: finish transform, accumulate penalty =========
  int carry2 = 1 << 20;
  int psum = 0, pcnt = 0;
  for (int k = NH / 2 - 1; k >= 0; --k) {
    const unsigned v2 = dist32[k * TPB + tid];
    const unsigned vh = v2 >> 16, vl = v2 & 0xFFFFu;
    int a = (int)(vh & 0x7FFu);
    int d = min(a, carry2 + 1);
    carry2 = d;
    if (vh & 0x8000u) { psum += d; pcnt += (d != 0); }
    a = (int)(vl & 0x7FFu);
    d = min(a, carry2 + 1);
    carry2 = d;
    if (vl & 0x8000u) { psum += d; pcnt += (d != 0); }
  }

  // ================= block reduction -> global atomics =========================
  if (tid == 0) { s_red.bce = 0.0f; s_red.tot = 0.0f; s_red.cnt = 0u; }
  __syncthreads();
  atomicAdd(&s_red.bce, bce);
  atomicAdd(&s_red.tot, (float)psum);
  atomicAdd(&s_red.cnt, (unsigned)pcnt);
  __syncthreads();
  if (tid == 0) {
    atomicAdd(&acc[0], s_red.bce);
    atomicAdd(&acc[1], s_red.tot);
    atomicAdd(cnt, s_red.cnt);
  }
}

__global__ void dtl_init(float* acc, unsigned* cnt) {
  acc[0] = 0.0f; acc[1] = 0.0f; *cnt = 0u;
}

__global__ void dtl_final(const float* __restrict__ acc,
                          const unsigned* __restrict__ cnt,
                          float* __restrict__ out) {
  const float bce    = acc[0] / 33554432.0f;   // mean over N*C*H*W
  const float total  = acc[1];
  const float count  = (float)(*cnt);
  const float border = (total == 0.0f) ? 0.0f : (total / fmaxf(count, 1.0f));
  out[0] = bce + sqrtf(border);                // BORDER_WEIGHT=1, power=0.5
}

extern "C" void kernel_launch(void* const* d_in, const int* in_sizes, int n_in,
                              void* d_out, int out_size, void* d_ws, size_t ws_size,
                              hipStream_t stream) {
  const float* preds = (const float*)d_in[0];
  const float* tgts  = (const float*)d_in[1];
  float*    acc = (float*)d_ws;
  unsigned* cnt = (unsigned*)((char*)d_ws + 2 * sizeof(float));

  dtl_init<<<1, 1, 0, stream>>>(acc, cnt);
  dtl_main<<<(NIMG * NW) / TPB, TPB, LDS_TOTAL, stream>>>(preds, tgts, acc, cnt);
  dtl_final<<<1, 1, 0, stream>>>(acc, cnt, (float*)d_out);
}